// CompareFeaSt_65343632441453
// MI455X (gfx1250) — compile-verified
//
#include <hip/hip_runtime.h>
#include <hip/hip_bf16.h>

typedef __attribute__((ext_vector_type(2))) float v2f;
typedef __attribute__((ext_vector_type(4))) float v4f;
typedef __attribute__((ext_vector_type(8))) float v8f;

// ---------------------------------------------------------------------------
// Zero-fill
// ---------------------------------------------------------------------------
__global__ void fill0_kernel(float* __restrict__ p, long n) {
    long i = (long)blockIdx.x * blockDim.x + threadIdx.x;
    if (i < n) p[i] = 0.0f;
}

// ---------------------------------------------------------------------------
// Repack [W_h0 | W_h1 | Ws | u] into a zero-padded [kpad x zld] panel with
// fixed slot stride sp (cout padded to sp): h0 @ 0, h1 @ sp, Ws @ 2sp, u @ 3sp.
// ---------------------------------------------------------------------------
__global__ void repack_w_kernel(const float* __restrict__ W,
                                const float* __restrict__ u,
                                const float* __restrict__ Ws,
                                int cin, int cout, int sp, int zld,
                                float* __restrict__ Wr, long tot) {
    long idx = (long)blockIdx.x * blockDim.x + threadIdx.x;
    if (idx >= tot) return;
    int k = (int)(idx / zld);
    int n = (int)(idx % zld);
    float v = 0.0f;
    if (k < cin) {
        if (n < sp)          { if (n < cout) v = W[(size_t)k * (2 * cout) + n]; }
        else if (n < 2 * sp) { int c = n - sp;     if (c < cout) v = W[(size_t)k * (2 * cout) + cout + c]; }
        else if (n < 3 * sp) { int c = n - 2 * sp; if (c < cout) v = Ws[(size_t)k * cout + c]; }
        else                 { int c = n - 3 * sp; if (c < 2)    v = u[(size_t)k * 2 + c]; }
    }
    Wr[idx] = v;
}

// ---------------------------------------------------------------------------
// Build the 22-channel input, zero-padded to stride 32.
// ---------------------------------------------------------------------------
__global__ void build_input_kernel(const float* __restrict__ featr3,
                                   const float* __restrict__ stm,
                                   int n, float* __restrict__ out) {
    const int rows[6] = {0, 0, 0, 1, 1, 2};
    const int cols[6] = {0, 1, 2, 1, 2, 2};
    long idx = (long)blockIdx.x * blockDim.x + threadIdx.x;
    if (idx >= (long)n * 32) return;
    int i = (int)(idx >> 5);
    int c = (int)(idx & 31);
    float v = 0.0f;
    if (c < 6)        v = featr3[(size_t)i * 27 + 0 * 9 + rows[c] * 3 + cols[c]];
    else if (c < 12)  v = featr3[(size_t)i * 27 + 1 * 9 + rows[c - 6] * 3 + cols[c - 6]];
    else if (c < 21)  v = featr3[(size_t)i * 27 + 2 * 9 + (c - 12)];
    else if (c == 21) v = stm[i];
    out[idx] = v;
}

// ---------------------------------------------------------------------------
// Node GEMM via V_WMMA_F32_16X16X4_F32:  Z[n, zld] = X[n, ld_in] @ Wr[kpad, zld]
// One wave per 16x16 output tile; 8 waves per 256-thread block.
// A frag (16x4 f32): lanes 0-15 K={k,k+1}, lanes 16-31 K={k+2,k+3} (one b64).
// B frag mirrored. C/D: 8 VGPRs, M = v + 8*(lane>=16), N = lane&15.
// Out-of-range rows are clamped (loads always legal) and discarded at store,
// so EXEC stays all-ones with zero per-load predication.
// ---------------------------------------------------------------------------
__global__ void __launch_bounds__(256)
gemm_wmma_kernel(const float* __restrict__ X, int n, int ld_in, int kpad,
                 const float* __restrict__ Wr, int zld,
                 float* __restrict__ Z) {
    const int lane  = threadIdx.x & 31;
    const int wave  = threadIdx.x >> 5;
    const int mbase = blockIdx.x * 16;
    const int ntile = blockIdx.y * 8 + wave;
    const int ntcnt = zld >> 4;
    if (ntile >= ntcnt) return;             // wave-uniform exit: EXEC stays full
    const int nbase = ntile * 16;
    const int r     = lane & 15;
    const int khi   = (lane >> 4) << 1;     // 0 or 2

    int row = mbase + r;
    if (row >= n) row = n - 1;              // clamp: discarded at store

    const float* ap = X + (size_t)row * ld_in + khi;            // 8B aligned
    const float* bp = Wr + (size_t)khi * zld + nbase + r;
    const size_t bstep = (size_t)4 * zld;

    v8f acc = {};
#pragma unroll 4
    for (int k = 0; k < kpad; k += 4) {
        v2f a = *(const v2f*)ap;            // global_load_b64
        v2f b;
        b.x = bp[0];
        b.y = bp[zld];
        __builtin_prefetch(bp + 2 * bstep, 0, 1);
        acc = __builtin_amdgcn_wmma_f32_16x16x4_f32(
            false, a, false, b, (short)0, acc, false, false);
        ap += 4;
        bp += bstep;
    }

    const int rowoff = (lane >> 4) << 3;    // 0 or 8
#pragma unroll
    for (int v = 0; v < 8; ++v) {
        int rr = mbase + rowoff + v;
        if (rr < n) Z[(size_t)rr * zld + nbase + r] = acc[v];
    }
}

// ---------------------------------------------------------------------------
// Edge phase: one wave per edge; branch-free vectorized scatter.
// q = softmax over 2 heads of (x@u)[s]-(x@u)[d]+c (u columns at Z[:, 3*sp..]);
// each lane blends 4 consecutive columns via two b128 loads and scatters with
// float atomics. Slot padding is zero, so pad columns add exact zeros.
// ---------------------------------------------------------------------------
__global__ void __launch_bounds__(256)
edge_kernel(const int* __restrict__ ei, int E,
            const float* __restrict__ Z, int zld, int sp,
            const float* __restrict__ cvec,
            float* __restrict__ agg, float* __restrict__ deg) {
    int e = blockIdx.x * 8 + (threadIdx.x >> 5);
    if (e >= E) return;
    const int lane = threadIdx.x & 31;
    const int s = ei[e];
    const int d = ei[(size_t)E + e];
    const float* zs = Z + (size_t)s * zld;
    const float* zd = Z + (size_t)d * zld;
    const int uoff = 3 * sp;
    const float t0 = zs[uoff]     - zd[uoff]     + cvec[0];
    const float t1 = zs[uoff + 1] - zd[uoff + 1] + cvec[1];
    const float m  = fmaxf(t0, t1);
    const float e0 = __expf(t0 - m);
    const float e1 = __expf(t1 - m);
    const float inv = 1.0f / (e0 + e1);
    const float q0 = e0 * inv, q1 = e1 * inv;
    float* ag = agg + (size_t)d * 128;
    const int c4 = lane * 4;
    if (c4 < sp) {
        v4f a = *(const v4f*)(zs + c4);          // head0, 16B aligned
        v4f b = *(const v4f*)(zs + sp + c4);     // head1, 16B aligned
        atomicAdd(&ag[c4 + 0], q0 * a.x + q1 * b.x);
        atomicAdd(&ag[c4 + 1], q0 * a.y + q1 * b.y);
        atomicAdd(&ag[c4 + 2], q0 * a.z + q1 * b.z);
        atomicAdd(&ag[c4 + 3], q0 * a.w + q1 * b.w);
    }
    if (lane == 0) atomicAdd(&deg[d], 1.0f);
}

// ---------------------------------------------------------------------------
// Finalize: y = agg/max(deg,1) + b + (x@Ws) + bs, optional ReLU.
// Writes zeros into pad columns so the next GEMM needs no K-mask.
// ---------------------------------------------------------------------------
__global__ void finalize_kernel(int n, int cout, int sp, int zld,
                                const float* __restrict__ Z,
                                const float* __restrict__ agg,
                                const float* __restrict__ deg,
                                const float* __restrict__ b,
                                const float* __restrict__ bs,
                                int relu, float* __restrict__ out, int ld_out) {
    long idx = (long)blockIdx.x * blockDim.x + threadIdx.x;
    if (idx >= (long)n * ld_out) return;
    int i = (int)(idx / ld_out);
    int c = (int)(idx % ld_out);
    float y = 0.0f;
    if (c < cout) {
        float dg = fmaxf(deg[i], 1.0f);
        y = agg[(size_t)i * 128 + c] / dg + b[c]
          + Z[(size_t)i * zld + 2 * sp + c] + bs[c];
        if (relu) y = fmaxf(y, 0.0f);
    }
    out[idx] = y;
}

// ---------------------------------------------------------------------------
// Cluster mean-pool: scatter (wave per fine node) then divide.
// ---------------------------------------------------------------------------
__global__ void __launch_bounds__(256)
pool_scatter_kernel(const float* __restrict__ X, int ldx, int n_fine,
                    const int* __restrict__ cluster,
                    float* __restrict__ dst, float* __restrict__ cnt) {
    int i = blockIdx.x * 8 + (threadIdx.x >> 5);
    if (i >= n_fine) return;
    const int lane = threadIdx.x & 31;
    const int j = cluster[i];
    const int c4 = lane * 4;
    if (c4 < 116) {                          // covers 115 cols; col 115 pad is 0
        v4f a = *(const v4f*)(X + (size_t)i * ldx + c4);
        atomicAdd(&dst[(size_t)j * 128 + c4 + 0], a.x);
        atomicAdd(&dst[(size_t)j * 128 + c4 + 1], a.y);
        atomicAdd(&dst[(size_t)j * 128 + c4 + 2], a.z);
        atomicAdd(&dst[(size_t)j * 128 + c4 + 3], a.w);
    }
    if (lane == 0) atomicAdd(&cnt[j], 1.0f);
}

__global__ void pool_div_kernel(float* __restrict__ dst,
                                const float* __restrict__ cnt, int n_coarse) {
    long idx = (long)blockIdx.x * blockDim.x + threadIdx.x;
    if (idx >= (long)n_coarse * 128) return;
    int i = (int)(idx >> 7);
    int c = (int)(idx & 127);
    dst[idx] = (c < 115) ? dst[idx] / fmaxf(cnt[i], 1.0f) : 0.0f;
}

// ---------------------------------------------------------------------------
// Unpool (gather coarse by cluster) + concat; [x_unpooled | copy | zero pad].
// ---------------------------------------------------------------------------
__global__ void unpool_concat_kernel(const float* __restrict__ Xc,
                                     const int* __restrict__ cluster,
                                     const float* __restrict__ cpy,
                                     int n_fine, float* __restrict__ out) {
    long idx = (long)blockIdx.x * blockDim.x + threadIdx.x;
    if (idx >= (long)n_fine * 256) return;
    int i = (int)(idx >> 8);
    int c = (int)(idx & 255);
    float v = 0.0f;
    if (c < 115)       v = Xc[(size_t)cluster[i] * 128 + c];
    else if (c < 230)  v = cpy[(size_t)i * 128 + (c - 115)];
    out[idx] = v;
}

// ---------------------------------------------------------------------------
// Host orchestration
// ---------------------------------------------------------------------------
extern "C" void kernel_launch(void* const* d_in, const int* in_sizes, int n_in,
                              void* d_out, int out_size, void* d_ws, size_t ws_size,
                              hipStream_t stream) {
    (void)n_in; (void)out_size; (void)ws_size;
    const float* featr3  = (const float*)d_in[0];
    const float* stmdist = (const float*)d_in[1];
    const int*   ei0 = (const int*)d_in[86];
    const int*   ei1 = (const int*)d_in[87];
    const int*   ei2 = (const int*)d_in[88];
    const int*   cl1 = (const int*)d_in[89];
    const int*   cl2 = (const int*)d_in[90];
    const int n0 = in_sizes[1];          // stmdist is [N0,1]
    const int n1 = in_sizes[90];         // cluster2 is [N1]
    const int n2 = 3125;
    const int E0 = in_sizes[86] / 2;
    const int E1 = in_sizes[87] / 2;
    const int E2 = in_sizes[88] / 2;

    static const int CIN[14]  = {22,115,115,115,115,115,230,115,115,115,230,115,115,115};
    static const int COUT[14] = {115,115,115,115,115,115,115,115,115,115,115,115,115,3};
    static const int RELU[14] = {1,1,1,1,1,1,1,1,1,1,1,1,1,0};
    int KP[14], SP[14], ZLD[14];
    float* WREP[14];

    float* ws = (float*)d_ws;
    size_t off = 0;
    auto take = [&](size_t nf) { float* p = ws + off; off += nf; return p; };

    for (int i = 0; i < 14; ++i) {
        KP[i]   = ((CIN[i] + 15) / 16) * 16;
        SP[i]   = ((COUT[i] + 15) / 16) * 16;     // 128 for 115, 16 for 3
        ZLD[i]  = 3 * SP[i] + 16;                  // 400 / 64
        WREP[i] = take((size_t)KP[i] * ZLD[i]);
    }
    float* Xin   = take((size_t)n0 * 32);
    float* X0a   = take((size_t)n0 * 128);
    float* X0b   = take((size_t)n0 * 128);
    float* X0cat = take((size_t)n0 * 256);
    float* C0    = take((size_t)n0 * 128);
    float* X1a   = take((size_t)n1 * 128);
    float* X1b   = take((size_t)n1 * 128);
    float* X1cat = take((size_t)n1 * 256);
    float* C1    = take((size_t)n1 * 128);
    float* X2a   = take((size_t)n2 * 128);
    float* X2b   = take((size_t)n2 * 128);
    float* Zbuf  = take((size_t)n0 * 400);
    float* AGG   = take((size_t)n0 * 128);
    float* DEG   = take((size_t)n0);

    // Repack weights for all 14 blocks.
    for (int i = 0; i < 14; ++i) {
        const float* W   = (const float*)d_in[2 + 6 * i + 0];
        const float* u   = (const float*)d_in[2 + 6 * i + 1];
        const float* Wsm = (const float*)d_in[2 + 6 * i + 4];
        long tot = (long)KP[i] * ZLD[i];
        repack_w_kernel<<<(unsigned)((tot + 255) / 256), 256, 0, stream>>>(
            W, u, Wsm, CIN[i], COUT[i], SP[i], ZLD[i], WREP[i], tot);
    }

    // Input assembly.
    {
        long tot = (long)n0 * 32;
        build_input_kernel<<<(unsigned)((tot + 255) / 256), 256, 0, stream>>>(
            featr3, stmdist, n0, Xin);
    }

    auto run_block = [&](int bi, const float* xin, int ld_in, int n,
                         const int* ei, int E, float* xout, int ld_out) {
        const int cout = COUT[bi], sp = SP[bi], zld = ZLD[bi], kpad = KP[bi];
        const float* cvec = (const float*)d_in[2 + 6 * bi + 2];
        const float* bb   = (const float*)d_in[2 + 6 * bi + 3];
        const float* bsv  = (const float*)d_in[2 + 6 * bi + 5];
        fill0_kernel<<<(unsigned)(((long)n * 128 + 255) / 256), 256, 0, stream>>>(AGG, (long)n * 128);
        fill0_kernel<<<(unsigned)((n + 255) / 256), 256, 0, stream>>>(DEG, (long)n);
        dim3 g((unsigned)((n + 15) / 16), (unsigned)((zld / 16 + 7) / 8));
        gemm_wmma_kernel<<<g, 256, 0, stream>>>(xin, n, ld_in, kpad, WREP[bi], zld, Zbuf);
        edge_kernel<<<(unsigned)((E + 7) / 8), 256, 0, stream>>>(ei, E, Zbuf, zld, sp, cvec, AGG, DEG);
        long tot = (long)n * ld_out;
        finalize_kernel<<<(unsigned)((tot + 255) / 256), 256, 0, stream>>>(
            n, cout, sp, zld, Zbuf, AGG, DEG, bb, bsv, RELU[bi], xout, ld_out);
    };

    auto run_pool = [&](const float* xin, int n_fine, const int* cluster,
                        float* xout, int n_coarse) {
        fill0_kernel<<<(unsigned)(((long)n_coarse * 128 + 255) / 256), 256, 0, stream>>>(
            xout, (long)n_coarse * 128);
        fill0_kernel<<<(unsigned)((n_coarse + 255) / 256), 256, 0, stream>>>(DEG, (long)n_coarse);
        pool_scatter_kernel<<<(unsigned)((n_fine + 7) / 8), 256, 0, stream>>>(
            xin, 128, n_fine, cluster, xout, DEG);
        pool_div_kernel<<<(unsigned)(((long)n_coarse * 128 + 255) / 256), 256, 0, stream>>>(
            xout, DEG, n_coarse);
    };

    run_block(0, Xin, 32, n0, ei0, E0, X0a, 128);
    run_block(1, X0a, 128, n0, ei0, E0, C0, 128);        // copy0
    run_pool(C0, n0, cl1, X1a, n1);
    run_block(2, X1a, 128, n1, ei1, E1, X1b, 128);
    run_block(3, X1b, 128, n1, ei1, E1, C1, 128);        // copy1
    run_pool(C1, n1, cl2, X2a, n2);
    run_block(4, X2a, 128, n2, ei2, E2, X2b, 128);
    run_block(5, X2b, 128, n2, ei2, E2, X2a, 128);
    unpool_concat_kernel<<<(unsigned)(((long)n1 * 256 + 255) / 256), 256, 0, stream>>>(
        X2a, cl2, C1, n1, X1cat);
    run_block(6, X1cat, 256, n1, ei1, E1, X1a, 128);
    run_block(7, X1a, 128, n1, ei1, E1, X1b, 128);
    run_block(8, X1b, 128, n1, ei1, E1, X1a, 128);
    run_block(9, X1a, 128, n1, ei1, E1, X1b, 128);
    unpool_concat_kernel<<<(unsigned)(((long)n0 * 256 + 255) / 256), 256, 0, stream>>>(
        X1b, cl1, C0, n0, X0cat);
    run_block(10, X0cat, 256, n0, ei0, E0, X0a, 128);
    run_block(11, X0a, 128, n0, ei0, E0, X0b, 128);
    run_block(12, X0b, 128, n0, ei0, E0, X0a, 128);
    run_block(13, X0a, 128, n0, ei0, E0, (float*)d_out, 3);
}